// WOS_55576876810248
// MI455X (gfx1250) — compile-verified
//
#include <hip/hip_runtime.h>
#include <cstdint>
#include <math.h>

// Problem constants (match reference)
#define Bn   4
#define Cn   3
#define Hn   224
#define Wn   224
#define On   8
#define Dn   27          // C*K*K
#define TS   16          // spatial tile (224 = 14*16, no partial tiles)
#define TP   (TS + 2)    // tile + 1-px halo on each side
#define TILE_ELEMS (Cn * TP * TP)   // 972 floats = 3888 B

// Weighted-order-statistic kernel.
// grid = (14*14 tiles, O, B), block = (16,16) = 256 threads = 8 wave32 waves.
__global__ __launch_bounds__(256)
void wos_kernel(const float* __restrict__ x,
                const float* __restrict__ mask,
                const float* __restrict__ weight,
                const float* __restrict__ bias,
                float* __restrict__ out)
{
    __shared__ float tile[TILE_ELEMS];        // zero-padded input tile (C,18,18)
    __shared__ float zsh[Dn * 256];           // per-lane z columns for dynamic index

    const int tid = threadIdx.y * TS + threadIdx.x;
    const int o   = blockIdx.y;
    const int b   = blockIdx.z;
    const int tY  = blockIdx.x / (Wn / TS);
    const int tX  = blockIdx.x % (Wn / TS);
    const int h0  = tY * TS;
    const int w0  = tX * TS;

    // ---- 1) zero-fill tile so out-of-bounds halo reads 'same' padding ----
    for (int e = tid; e < TILE_ELEMS; e += 256) tile[e] = 0.0f;
    __syncthreads();   // drains DScnt before the async engine writes LDS

    // ---- 2) async DMA of in-bounds tile elements: global -> LDS ----------
    // CDNA5 path: GLOBAL_LOAD_ASYNC_TO_LDS_B32 (ASYNCcnt), EXEC-masked so
    // out-of-bounds lanes keep the zero padding.
    const float* xb = x + (size_t)b * (Cn * Hn * Wn);
    for (int e = tid; e < TILE_ELEMS; e += 256) {
        const int c  = e / (TP * TP);
        const int r  = (e / TP) % TP;
        const int s  = e % TP;
        const int gh = h0 - 1 + r;
        const int gw = w0 - 1 + s;
        if (gh >= 0 && gh < Hn && gw >= 0 && gw < Wn) {
            const uint32_t voff = (uint32_t)(((c * Hn + gh) * Wn + gw) * 4);
            const uint32_t lds  = (uint32_t)(uintptr_t)(&tile[e]);
            asm volatile("global_load_async_to_lds_b32 %0, %1, %2"
                         :: "v"(lds), "v"(voff), "s"(xb) : "memory");
        }
    }
    asm volatile("s_wait_asynccnt 0" ::: "memory");
    __syncthreads();

    // ---- 3) build z[k] = patch[k] + mask[o][k] in registers --------------
    // torch-unfold ordering: k = c*9 + dy*3 + dx
    const float* mo = mask   + o * Dn;
    const float* wo = weight + o * 2 * Dn;
    const float  bo = bias[o];

    float z[Dn];
    #pragma unroll
    for (int c = 0; c < Cn; ++c)
        #pragma unroll
        for (int dy = 0; dy < 3; ++dy)
            #pragma unroll
            for (int dx = 0; dx < 3; ++dx) {
                const int k = c * 9 + dy * 3 + dx;
                const float v = tile[(c * TP + threadIdx.y + dy) * TP
                                     + threadIdx.x + dx];
                z[k] = v + mo[k];    // mo uniform -> scalar loads
            }

    float wp[Dn], wm[Dn];
    #pragma unroll
    for (int k = 0; k < Dn; ++k) { wp[k] = wo[k]; wm[k] = wo[Dn + k]; }

    // stash z in per-lane LDS columns (stride 256 -> conflict-free) so the
    // outer selection loop can index it dynamically without scratch spills
    #pragma unroll
    for (int k = 0; k < Dn; ++k) zsh[k * 256 + tid] = z[k];

    // ---- 4) sort-free weighted order-statistic selection -----------------
    // For each of the 54 elements {+z_c, -z_c}: cum = inclusive cumulative
    // weight at its sorted position (masked sums, ties = stable argsort with
    // the +z half first). Answer = min value with cum <= bias, else max.
    // Four parity-split accumulators per element pair -> 4 independent
    // cmp/cndmask/add chains for VOPD dual-issue; halves the FP dep chain.
    float ymin = INFINITY;
    float ymax = -INFINITY;

    #pragma unroll 1
    for (int c = 0; c < Dn; ++c) {
        const float zc  = zsh[c * 256 + tid];
        const float nzc = -zc;
        float cpA = 0.0f, cpB = 0.0f;   // cum of element +z_c (split)
        float cmA = 0.0f, cmB = 0.0f;   // cum of element -z_c (split)
        #pragma unroll
        for (int i = 0; i < Dn; i += 2) {
            {
                const float zi = z[i];
                cpA += (zi >= zc)  ? wp[i] : 0.0f;   // +z_i at/above +z_c
                cpB += (zi <  nzc) ? wm[i] : 0.0f;   // -z_i strictly above +z_c
                cmA += (zi >= nzc) ? wp[i] : 0.0f;   // +z_i at/above -z_c (tie: + first)
                cmB += (zi <= zc)  ? wm[i] : 0.0f;   // -z_i at/above -z_c (incl. self)
            }
            if (i + 1 < Dn) {
                const float zi = z[i + 1];
                cpB += (zi >= zc)  ? wp[i + 1] : 0.0f;
                cpA += (zi <  nzc) ? wm[i + 1] : 0.0f;
                cmB += (zi >= nzc) ? wp[i + 1] : 0.0f;
                cmA += (zi <= zc)  ? wm[i + 1] : 0.0f;
            }
        }
        const float cp = cpA + cpB;
        const float cm = cmA + cmB;
        if (cp <= bo) ymin = fminf(ymin, zc);
        if (cm <= bo) ymin = fminf(ymin, nzc);
        ymax = fmaxf(ymax, fmaxf(zc, nzc));
    }
    const float y = (ymin != INFINITY) ? ymin : ymax;   // li clipped to 0 case

    const int h = h0 + threadIdx.y;
    const int w = w0 + threadIdx.x;
    out[(((size_t)b * On + o) * Hn + h) * Wn + w] = y;  // coalesced store
}

extern "C" void kernel_launch(void* const* d_in, const int* in_sizes, int n_in,
                              void* d_out, int out_size, void* d_ws, size_t ws_size,
                              hipStream_t stream)
{
    const float* x      = (const float*)d_in[0];
    const float* mask   = (const float*)d_in[1];
    const float* weight = (const float*)d_in[2];
    const float* bias   = (const float*)d_in[3];
    float* out          = (float*)d_out;

    dim3 grid((Hn / TS) * (Wn / TS), On, Bn);   // (196, 8, 4)
    dim3 block(TS, TS, 1);                       // 256 threads = 8 waves
    hipLaunchKernelGGL(wos_kernel, grid, block, 0, stream,
                       x, mask, weight, bias, out);
}